// MultiDomainHypergraphEncoder_29978871726112
// MI455X (gfx1250) — compile-verified
//
#include <hip/hip_runtime.h>
#include <hip/hip_bf16.h>
#include <math.h>

// ---------------------------------------------------------------------------
// MultiDomainHypergraphEncoder for MI455X (gfx1250, wave32)
// BS=32, SEQ=512, ENC=8, D=128, NFREQ=32, F=257
// ---------------------------------------------------------------------------

#define BS   32
#define SEQ  512
#define ENC  8
#define DM   128
#define NFQ  32
#define NF   257   // rfft bins = SEQ/2 + 1

typedef __attribute__((ext_vector_type(2))) float v2f;
typedef __attribute__((ext_vector_type(8))) float v8f;

// ----- output layout (floats), in reference return order --------------------
#define O_TIME   ((size_t)0)                       // (32,512,8,128)  16777216
#define O_FREQ   (O_TIME + (size_t)BS*SEQ*ENC*DM)  // (32,32,8,128)    1048576
#define O_STAT   (O_FREQ + (size_t)BS*NFQ*ENC*DM)  // (32,3,8,128)       98304
#define O_TH     (O_STAT + (size_t)BS*3*ENC*DM)    // (32,512,128)     2097152
#define O_FH     (O_TH   + (size_t)BS*SEQ*DM)      // (32,32,128)       131072
#define O_CD     (O_FH   + (size_t)BS*NFQ*DM)      // (32,3,128)         12288
#define O_INC    (O_CD   + (size_t)BS*3*DM)        // (32,512,4096)   67108864

// ----- workspace layout (floats) --------------------------------------------
#define W_TMEAN  ((size_t)0)                        // (32,512,128)
#define W_XR     (W_TMEAN + (size_t)BS*SEQ*DM)      // (32,257,8)
#define W_XI     (W_XR    + (size_t)BS*NF*ENC)      // (32,257,8)
#define W_MAG    (W_XI    + (size_t)BS*NF*ENC)      // (32,257)
#define W_TOPK   (W_MAG   + (size_t)BS*NF)          // (32,32) ints
#define W_FMEAN  (W_TOPK  + (size_t)BS*NFQ)         // (32,32,128)
#define W_STATF  (W_FMEAN + (size_t)BS*NFQ*DM)      // (32,8,3)
#define W_STATV  (W_STATF + (size_t)BS*ENC*3)       // (32,8,128)
#define W_CROSS  (W_STATV + (size_t)BS*ENC*DM)      // (32,3,128)

// ===========================================================================
// K1: time_nodes = relu(x[...,None]*W_time + b_time); also per-(b,s) enc-mean
// ===========================================================================
__global__ __launch_bounds__(DM)
void k_time_nodes(const float* __restrict__ x, const float* __restrict__ Wt,
                  const float* __restrict__ bt, float* __restrict__ out_time,
                  float* __restrict__ tmean) {
  const int bsm = blockIdx.x;          // b*512+s
  const int d   = threadIdx.x;
  const float w = Wt[d], bb = bt[d];
  const float* xp = x + (size_t)bsm * ENC;
  float* op = out_time + (size_t)bsm * ENC * DM + d;
  float acc = 0.f;
#pragma unroll
  for (int e = 0; e < ENC; ++e) {
    float v = fmaxf(xp[e] * w + bb, 0.f);
    op[(size_t)e * DM] = v;
    acc += v;
  }
  tmean[(size_t)bsm * DM + d] = acc * 0.125f;
}

// ===========================================================================
// K2: naive DFT (rfft): Xr/Xi[b,f,e], mag_mean[b,f] = mean_e |X|
// block = (f, b), 64 threads: e = t&7, 8 chunks of 64 samples
// ===========================================================================
__global__ __launch_bounds__(64)
void k_dft(const float* __restrict__ x, float* __restrict__ Xr,
           float* __restrict__ Xi, float* __restrict__ mag) {
  const int f = blockIdx.x, b = blockIdx.y;
  const int t = threadIdx.x, e = t & 7, part = t >> 3;
  const float w0 = 6.2831853071795864769f / (float)SEQ;
  float re = 0.f, im = 0.f;
  for (int i = 0; i < SEQ / 8; ++i) {
    int s = part * (SEQ / 8) + i;
    float xv = x[((size_t)(b * SEQ + s)) * ENC + e];
    float th = (float)((f * s) & (SEQ - 1)) * w0;   // reduced angle
    float sn, cs;
    sincosf(th, &sn, &cs);
    re += xv * cs;
    im -= xv * sn;
  }
  __shared__ float sre[64], sim[64];
  sre[t] = re; sim[t] = im;
  __syncthreads();
  if (t < ENC) {
    float R = 0.f, I = 0.f;
#pragma unroll
    for (int p = 0; p < 8; ++p) { R += sre[p * 8 + t]; I += sim[p * 8 + t]; }
    size_t o = ((size_t)b * NF + f) * ENC + t;
    Xr[o] = R; Xi[o] = I;
    sre[t] = sqrtf(R * R + I * I);
  }
  __syncthreads();
  if (t == 0) {
    float m = 0.f;
#pragma unroll
    for (int e2 = 0; e2 < ENC; ++e2) m += sre[e2];
    mag[(size_t)b * NF + f] = m * 0.125f;
  }
}

// ===========================================================================
// K3: top-32 of 257 magnitudes per batch (descending, lower index on ties)
// ===========================================================================
__global__ __launch_bounds__(256)
void k_topk(const float* __restrict__ mag, int* __restrict__ topk) {
  const int b = blockIdx.x, t = threadIdx.x;
  __shared__ float sv[NF];
  __shared__ float rv[256];
  __shared__ int   ri[256];
  sv[t] = mag[(size_t)b * NF + t];
  if (t == 0) sv[256] = mag[(size_t)b * NF + 256];
  __syncthreads();
  for (int it = 0; it < NFQ; ++it) {
    float v = sv[t]; int idx = t;
    if (t == 0) {
      float v2 = sv[256];
      if (v2 > v) { v = v2; idx = 256; }   // tie keeps idx 0 (lower index)
    }
    rv[t] = v; ri[t] = idx;
    __syncthreads();
    for (int off = 128; off > 0; off >>= 1) {
      if (t < off) {
        float vo = rv[t + off]; int io = ri[t + off];
        if (vo > rv[t] || (vo == rv[t] && io < ri[t])) { rv[t] = vo; ri[t] = io; }
      }
      __syncthreads();
    }
    if (t == 0) { topk[b * NFQ + it] = ri[0]; sv[ri[0]] = -3.0e38f; }
    __syncthreads();
  }
}

// ===========================================================================
// K4: freq_nodes = relu(re*Wf[0]+im*Wf[1]+bf); also per-(b,k) enc-mean
// ===========================================================================
__global__ __launch_bounds__(DM)
void k_freq_nodes(const float* __restrict__ Xr, const float* __restrict__ Xi,
                  const int* __restrict__ topk, const float* __restrict__ Wf,
                  const float* __restrict__ bf, float* __restrict__ out_freq,
                  float* __restrict__ fmean) {
  const int bk = blockIdx.x;           // b*32+k
  const int b = bk >> 5;
  const int d = threadIdx.x;
  const int f = topk[bk];
  const float w0 = Wf[d], w1 = Wf[DM + d], bb = bf[d];
  const float* rp = Xr + ((size_t)b * NF + f) * ENC;
  const float* ip = Xi + ((size_t)b * NF + f) * ENC;
  float* op = out_freq + (size_t)bk * ENC * DM + d;
  float acc = 0.f;
#pragma unroll
  for (int e = 0; e < ENC; ++e) {
    float v = fmaxf(rp[e] * w0 + ip[e] * w1 + bb, 0.f);
    op[(size_t)e * DM] = v;
    acc += v;
  }
  fmean[(size_t)bk * DM + d] = acc * 0.125f;
}

// ===========================================================================
// K5: masked statistics per (b,e): mean / var / trend
// ===========================================================================
__global__ __launch_bounds__(256)
void k_stats(const float* __restrict__ x, const float* __restrict__ mask,
             float* __restrict__ statf) {
  const int be = blockIdx.x;           // b*8+e
  const int b = be >> 3, e = be & 7;
  const int t = threadIdx.x;
  float s0 = 0.f, s1 = 0.f, s2 = 0.f, s3 = 0.f;
  for (int s = t; s < SEQ; s += 256) {
    size_t o = ((size_t)(b * SEQ + s)) * ENC + e;
    float m = mask[o];
    float xm = x[o] * m;
    s0 += m; s1 += xm; s2 += xm * xm; s3 += xm * (float)s;
  }
  __shared__ float4 red[256];
  red[t] = make_float4(s0, s1, s2, s3);
  __syncthreads();
  for (int off = 128; off > 0; off >>= 1) {
    if (t < off) {
      red[t].x += red[t + off].x; red[t].y += red[t + off].y;
      red[t].z += red[t + off].z; red[t].w += red[t + off].w;
    }
    __syncthreads();
  }
  if (t == 0) {
    float n    = fmaxf(red[0].x, 1.0f);
    float mean = red[0].y / n;
    float var  = (red[0].z - 2.f * mean * red[0].y + mean * mean * red[0].x) / n;
    float trnd = red[0].w / n - mean * ((float)(SEQ - 1) * 0.5f);
    statf[be * 3 + 0] = mean;
    statf[be * 3 + 1] = var;
    statf[be * 3 + 2] = trnd;
  }
}

// ===========================================================================
// K6: stat_nodes = relu(feat @ W_stat + b_stat), broadcast over r=0..2
// ===========================================================================
__global__ __launch_bounds__(DM)
void k_stat_nodes(const float* __restrict__ statf, const float* __restrict__ Ws,
                  const float* __restrict__ bsb, float* __restrict__ out_stat,
                  float* __restrict__ statv) {
  const int be = blockIdx.x;           // b*8+e
  const int b = be >> 3, e = be & 7;
  const int d = threadIdx.x;
  float m = statf[be * 3], va = statf[be * 3 + 1], tr = statf[be * 3 + 2];
  float v = fmaxf(m * Ws[d] + va * Ws[DM + d] + tr * Ws[2 * DM + d] + bsb[d], 0.f);
#pragma unroll
  for (int r = 0; r < 3; ++r)
    out_stat[((size_t)(b * 3 + r) * ENC + e) * DM + d] = v;
  statv[(size_t)be * DM + d] = v;
}

// ===========================================================================
// K7: cross_in[b,0..2,d] = means over (s,e) of time / freq / stat nodes
// ===========================================================================
__global__ __launch_bounds__(DM)
void k_cross(const float* __restrict__ tmean, const float* __restrict__ fmean,
             const float* __restrict__ statv, float* __restrict__ cross_in) {
  const int b = blockIdx.x, d = threadIdx.x;
  float s = 0.f;
  for (int i = 0; i < SEQ; ++i) s += tmean[((size_t)b * SEQ + i) * DM + d];
  cross_in[((size_t)b * 3 + 0) * DM + d] = s * (1.f / (float)SEQ);
  s = 0.f;
  for (int i = 0; i < NFQ; ++i) s += fmean[((size_t)b * NFQ + i) * DM + d];
  cross_in[((size_t)b * 3 + 1) * DM + d] = s * (1.f / (float)NFQ);
  s = 0.f;
  for (int i = 0; i < ENC; ++i) s += statv[((size_t)b * ENC + i) * DM + d];
  cross_in[((size_t)b * 3 + 2) * DM + d] = s * 0.125f;
}

// ===========================================================================
// K8: fp32 WMMA GEMM:  out[M,128] = A[M,128] @ W[128,128] + bias
//   block = 128 threads = 4 waves; each wave owns one 16x128 output tile.
//   W is transposed into LDS once per block -> contiguous ds_load_b64 B-frags.
//   Uses v_wmma_f32_16x16x4_f32 (exact fp32, matches reference precision).
//   A frag:  lanes 0-15 K=k0,k0+1 ; lanes 16-31 K=k0+2,k0+3 (ISA 16x4 layout)
//   C frag:  VGPR v -> M = v + 8*(lane>=16), N = lane%16
// ===========================================================================
__global__ __launch_bounds__(128)
void k_gemm_wmma(const float* __restrict__ A, const float* __restrict__ W,
                 const float* __restrict__ bias, float* __restrict__ out, int M) {
  __shared__ float wt[DM * DM];        // 64 KB: wt[n*128+k] = W[k*128+n]
  const int t = threadIdx.x;
  for (int i = t; i < DM * DM; i += 128) {
    int k = i >> 7, n = i & 127;
    wt[n * DM + k] = W[i];
  }
  __syncthreads();

  const int wave = t >> 5, lane = t & 31;
  const int m0 = (blockIdx.x * 4 + wave) * 16;
  if (m0 >= M) return;

  const int lcol = lane & 15;          // N within tile / M row within tile
  const int kh   = (lane >> 4) * 2;    // K sub-offset: 0 or 2

  v8f zero = {};
  v8f acc[8];
#pragma unroll
  for (int n = 0; n < 8; ++n) acc[n] = zero;

  const float* arow = A + (size_t)(m0 + lcol) * DM + kh;
  for (int k0 = 0; k0 < DM; k0 += 4) {
    v2f a = *(const v2f*)(arow + k0);                       // 8B-aligned
#pragma unroll
    for (int n = 0; n < 8; ++n) {
      v2f bf = *(const v2f*)&wt[(n * 16 + lcol) * DM + k0 + kh];
      acc[n] = __builtin_amdgcn_wmma_f32_16x16x4_f32(
          false, a, false, bf, (short)0, acc[n], false, false);
    }
  }

#pragma unroll
  for (int n = 0; n < 8; ++n) {
    float bv = bias[n * 16 + lcol];
#pragma unroll
    for (int v = 0; v < 8; ++v) {
      int row = m0 + v + (kh << 2);    // +8 for lanes 16-31
      out[(size_t)row * DM + n * 16 + lcol] = acc[n][v] + bv;
    }
  }
}

// ===========================================================================
// K9: temporal incidence — float4 zero-fill + block-diagonal mask scatter
//   inc[b,s, s2*8+e] = mask[b,s,e] * (s2==s).  268 MB store, fully vectorized.
// ===========================================================================
__global__ __launch_bounds__(256)
void k_incidence(const float* __restrict__ mask, float4* __restrict__ out) {
  const unsigned idx = blockIdx.x * 256u + threadIdx.x;   // < 16777216
  const int row = idx >> 10;           // 4096 cols / 4 = 1024 float4 per row
  const int col = (idx & 1023) << 2;
  const int b = row >> 9, s = row & 511;
  float4 v = make_float4(0.f, 0.f, 0.f, 0.f);
  if ((col >> 3) == s)
    v = *(const float4*)&mask[((size_t)(b * SEQ + s)) * ENC + (col & 7)];
  out[idx] = v;
}

// ===========================================================================
extern "C" void kernel_launch(void* const* d_in, const int* in_sizes, int n_in,
                              void* d_out, int out_size, void* d_ws, size_t ws_size,
                              hipStream_t stream) {
  (void)in_sizes; (void)n_in; (void)out_size; (void)ws_size;
  const float* x      = (const float*)d_in[0];
  const float* mask   = (const float*)d_in[1];
  // d_in[2] = n_freq_modes (fixed 32; shapes are compile-time constants here)
  const float* W_time = (const float*)d_in[3];
  const float* b_time = (const float*)d_in[4];
  const float* W_freq = (const float*)d_in[5];
  const float* b_freq = (const float*)d_in[6];
  const float* W_stat = (const float*)d_in[7];
  const float* b_stat = (const float*)d_in[8];
  const float* W_th   = (const float*)d_in[9];
  const float* b_th   = (const float*)d_in[10];
  const float* W_fh   = (const float*)d_in[11];
  const float* b_fh   = (const float*)d_in[12];
  const float* W_cd   = (const float*)d_in[13];
  const float* b_cd   = (const float*)d_in[14];

  float* out = (float*)d_out;
  float* ws  = (float*)d_ws;

  float* tmean  = ws + W_TMEAN;
  float* Xr     = ws + W_XR;
  float* Xi     = ws + W_XI;
  float* mag    = ws + W_MAG;
  int*   topk   = (int*)(ws + W_TOPK);
  float* fmean  = ws + W_FMEAN;
  float* statf  = ws + W_STATF;
  float* statv  = ws + W_STATV;
  float* crossi = ws + W_CROSS;

  // time_nodes + per-(b,s) enc-mean
  k_time_nodes<<<BS * SEQ, DM, 0, stream>>>(x, W_time, b_time, out + O_TIME, tmean);
  // DFT + magnitudes
  k_dft<<<dim3(NF, BS), 64, 0, stream>>>(x, Xr, Xi, mag);
  // top-32 modes per batch
  k_topk<<<BS, 256, 0, stream>>>(mag, topk);
  // freq_nodes + per-(b,k) enc-mean
  k_freq_nodes<<<BS * NFQ, DM, 0, stream>>>(Xr, Xi, topk, W_freq, b_freq,
                                            out + O_FREQ, fmean);
  // masked stats
  k_stats<<<BS * ENC, 256, 0, stream>>>(x, mask, statf);
  // stat_nodes (+ per-(b,e) values for the cross mean)
  k_stat_nodes<<<BS * ENC, DM, 0, stream>>>(statf, W_stat, b_stat,
                                            out + O_STAT, statv);
  // cross-domain input rows
  k_cross<<<BS, DM, 0, stream>>>(tmean, fmean, statv, crossi);

  // hyperedge GEMMs (fp32 WMMA)
  k_gemm_wmma<<<(BS * SEQ) / 64, 128, 0, stream>>>(tmean, W_th, b_th,
                                                   out + O_TH, BS * SEQ);
  k_gemm_wmma<<<(BS * NFQ) / 64, 128, 0, stream>>>(fmean, W_fh, b_fh,
                                                   out + O_FH, BS * NFQ);
  k_gemm_wmma<<<(BS * 3 + 63) / 64, 128, 0, stream>>>(crossi, W_cd, b_cd,
                                                      out + O_CD, BS * 3);

  // block-diagonal incidence (dominant store traffic)
  k_incidence<<<(BS * SEQ * SEQ * ENC / 4) / 256, 256, 0, stream>>>(
      mask, (float4*)(out + O_INC));
}